// CostBasedRouter_52338471469232
// MI455X (gfx1250) — compile-verified
//
#include <hip/hip_runtime.h>
#include <hip/hip_bf16.h>
#include <math.h>

// CDNA5 fp32 WMMA fragment types
typedef __attribute__((ext_vector_type(2))) float v2f;   // A/B frag: 16x4 / 4x16 f32
typedef __attribute__((ext_vector_type(8))) float v8f;   // C/D frag: 16x16 f32
typedef int vi4 __attribute__((vector_size(16)));        // b128 payload type for async DMA

#define S_LEN 2048
#define B_SZ  16
#define D_IN  1024
#define H_SZ  256
#define G3    768               // 3*H
#define M_ROWS (S_LEN * B_SZ)   // 32768

#define AS1 __attribute__((address_space(1)))
#define AS3 __attribute__((address_space(3)))

// ---- CDNA5 async copy: global -> LDS (ASYNCcnt-tracked, no VGPR round-trip) ----
__device__ __forceinline__ void async_copy_b128(const float* gptr, float* lptr) {
#if defined(__has_builtin) && __has_builtin(__builtin_amdgcn_global_load_async_to_lds_b128)
  __builtin_amdgcn_global_load_async_to_lds_b128(
      (AS1 vi4*)gptr, (AS3 vi4*)lptr, 0, 0);
#else
  unsigned lds_off = (unsigned)(uintptr_t)(AS3 void*)lptr;
  asm volatile("global_load_async_to_lds_b128 %0, %1, off"
               :: "v"(lds_off), "v"(gptr) : "memory");
#endif
}

__device__ __forceinline__ void wait_async0() {
#if defined(__has_builtin) && __has_builtin(__builtin_amdgcn_s_wait_asynccnt)
  __builtin_amdgcn_s_wait_asynccnt(0);
#else
  asm volatile("s_wait_asynccnt 0x0" ::: "memory");
#endif
}

// ---------------------------------------------------------------------------
// Kernel 1: xp = x @ W_ih^T + b_ih    (M=32768, K=1024, N=768)
// Grid (1024, 2): block tile 32(M) x 384(N); 256 threads = 8 waves (2M x 4N),
// each wave: 1 m-tile x 6 n-tiles = 6 WMMA accumulators (48 VGPRs).
// K-chunks of 16, double-buffered in LDS, filled by async global->LDS B128 DMA.
// x is read only 2x (268MB), W_ih streams from L2 -> ~16us of HBM traffic.
// ---------------------------------------------------------------------------
__global__ __launch_bounds__(256) void gemm_xproj(const float* __restrict__ x,
                                                  const float* __restrict__ Wih,
                                                  const float* __restrict__ bih,
                                                  float* __restrict__ xp) {
  __shared__ float As[2][32 * 16];    //  4 KB
  __shared__ float Bs[2][384 * 16];   // 48 KB

  const int tid   = threadIdx.x;
  const int lane  = tid & 31;
  const int wave  = tid >> 5;
  const int waveM = wave & 1;    // 0..1  (16 rows each)
  const int waveN = wave >> 1;   // 0..3  (96 cols each)
  const int mBase = blockIdx.x * 32;
  const int nBase = blockIdx.y * 384;

  const int ln    = lane & 15;
  const int khalf = (lane >> 4) * 2;   // 0 -> K0/K1 ; 2 -> K2/K3

  v8f acc[6] = {};

  // issue async fills of one K-chunk into buffer `buf`
  auto issue_chunk = [&](int buf, int k0) {
    if (tid < 128) {                       // A: 32x16 = 128 float4
      int row = tid >> 2, c4 = tid & 3;
      async_copy_b128(x + (size_t)(mBase + row) * D_IN + k0 + c4 * 4,
                      &As[buf][row * 16 + c4 * 4]);
    }
#pragma unroll
    for (int r = 0; r < 6; ++r) {          // B: 384x16 = 1536 float4
      int id  = tid + r * 256;
      int row = id >> 2, c4 = id & 3;
      async_copy_b128(Wih + (size_t)(nBase + row) * D_IN + k0 + c4 * 4,
                      &Bs[buf][row * 16 + c4 * 4]);
    }
  };

  issue_chunk(0, 0);
  wait_async0();
  __syncthreads();

  for (int chunk = 0; chunk < D_IN / 16; ++chunk) {
    const int cur = chunk & 1;
    if (chunk + 1 < D_IN / 16) issue_chunk(cur ^ 1, (chunk + 1) * 16);

    const float* Ab = &As[cur][0];
    const float* Bb = &Bs[cur][0];
#pragma unroll
    for (int kk = 0; kk < 16; kk += 4) {
      v2f a = *(const v2f*)(Ab + (waveM * 16 + ln) * 16 + kk + khalf);
      v2f b[6];
#pragma unroll
      for (int j = 0; j < 6; ++j)
        b[j] = *(const v2f*)(Bb + (waveN * 96 + j * 16 + ln) * 16 + kk + khalf);
#pragma unroll
      for (int j = 0; j < 6; ++j)
        acc[j] = __builtin_amdgcn_wmma_f32_16x16x4_f32(
            false, a, false, b[j], (short)0, acc[j], false, false);
    }

    wait_async0();     // next buffer's DMA landed (per-wave), then...
    __syncthreads();   // ...all waves done reading `cur` before it is refilled
  }

  // epilogue: D layout -> VGPR r: lanes0-15 M=r, lanes16-31 M=r+8; N = lane&15
  const int mhi = (lane >> 4) * 8;
#pragma unroll
  for (int j = 0; j < 6; ++j) {
    int col = nBase + waveN * 96 + j * 16 + ln;
    float bias = bih[col];
#pragma unroll
    for (int r = 0; r < 8; ++r) {
      int row = mBase + waveM * 16 + mhi + r;
      xp[(size_t)row * G3 + col] = acc[j][r] + bias;
    }
  }
}

// ---------------------------------------------------------------------------
// Kernel 2: sequential GRU scan. One persistent workgroup (512 thr = 16 waves).
// h (16x256) lives in LDS. Wave w owns hidden columns [16w,16w+16) of ALL three
// gates (tiles w, w+16, w+32 of the 768-wide hp), so the gate math is wave-local.
// Per step: 3 accumulators x 64 K4-WMMAs; W_hh streams from L2 (768KB resident).
// ---------------------------------------------------------------------------
__global__ __launch_bounds__(512) void gru_scan(const float* __restrict__ xp,
                                                const float* __restrict__ Whh,
                                                const float* __restrict__ bhh,
                                                float* __restrict__ hs) {
  __shared__ float hbuf[16 * 258];   // padded stride 258 to spread LDS banks

  const int tid  = threadIdx.x;
  const int lane = tid & 31;
  const int wave = tid >> 5;              // 0..15
  const int ln    = lane & 15;
  const int khalf = (lane >> 4) * 2;
  const int col   = wave * 16 + ln;       // hidden column this lane produces

  for (int i = tid; i < 16 * 258; i += 512) hbuf[i] = 0.0f;
  __syncthreads();

  // B-fragment row bases: B = W_hh^T, so B[k][n] = Whh[n*256 + k]
  const float* Wr = Whh + (size_t)(0 * H_SZ + col) * H_SZ;
  const float* Wz = Whh + (size_t)(1 * H_SZ + col) * H_SZ;
  const float* Wn = Whh + (size_t)(2 * H_SZ + col) * H_SZ;
  const float bhr = bhh[col], bhz = bhh[H_SZ + col], bhn = bhh[2 * H_SZ + col];

  for (int t = 0; t < S_LEN; ++t) {
    v8f ar = {}, az = {}, an = {};
#pragma unroll 4
    for (int k = 0; k < H_SZ; k += 4) {
      v2f a  = *(const v2f*)(hbuf + ln * 258 + k + khalf);
      v2f br = *(const v2f*)(Wr + k + khalf);
      v2f bz = *(const v2f*)(Wz + k + khalf);
      v2f bn = *(const v2f*)(Wn + k + khalf);
      ar = __builtin_amdgcn_wmma_f32_16x16x4_f32(false, a, false, br, (short)0, ar, false, false);
      az = __builtin_amdgcn_wmma_f32_16x16x4_f32(false, a, false, bz, (short)0, az, false, false);
      an = __builtin_amdgcn_wmma_f32_16x16x4_f32(false, a, false, bn, (short)0, an, false, false);
    }
    __syncthreads();   // all waves done reading h(t-1)

    const float* xpt = xp + (size_t)t * B_SZ * G3;
    const int bb = (lane >> 4) * 8;
#pragma unroll
    for (int r = 0; r < 8; ++r) {
      int batch = bb + r;
      float xr = xpt[batch * G3 + col];
      float xz = xpt[batch * G3 + H_SZ + col];
      float xn = xpt[batch * G3 + 2 * H_SZ + col];
      float hr = ar[r] + bhr, hz = az[r] + bhz, hn = an[r] + bhn;
      float rg = 1.0f / (1.0f + expf(-(xr + hr)));
      float zg = 1.0f / (1.0f + expf(-(xz + hz)));
      float ng = tanhf(xn + rg * hn);
      float hprev = hbuf[batch * 258 + col];
      float hnew  = (1.0f - zg) * ng + zg * hprev;
      hbuf[batch * 258 + col] = hnew;
      hs[((size_t)t * B_SZ + batch) * H_SZ + col] = hnew;
    }
    __syncthreads();   // h(t) committed before next step's reads
  }
}

// ---------------------------------------------------------------------------
// Kernel 3a: zero the segment-sum bins
// ---------------------------------------------------------------------------
__global__ void zero_bins(float* __restrict__ bins) {
  if (threadIdx.x < 16) bins[threadIdx.x] = 0.0f;
}

// ---------------------------------------------------------------------------
// Kernel 3b: router head: logits = relu(hs) @ W_out^T + b_out (E=4), softmax,
// usage = softmax . expert_dims, fused segment-sum via 8-bin atomics.
// ---------------------------------------------------------------------------
__global__ __launch_bounds__(256) void router_head(const float* __restrict__ hs,
                                                   const int* __restrict__ task_ids,
                                                   const float* __restrict__ Wout,
                                                   const float* __restrict__ bout,
                                                   float* __restrict__ out_raw,
                                                   float* __restrict__ out_router,
                                                   float* __restrict__ out_idx,
                                                   int idx_is_i64,
                                                   float* __restrict__ sums,
                                                   float* __restrict__ counts) {
  __shared__ float Wl[4 * H_SZ];
  __shared__ float bl[4];
  for (int i = threadIdx.x; i < 4 * H_SZ; i += 256) Wl[i] = Wout[i];
  if (threadIdx.x < 4) bl[threadIdx.x] = bout[threadIdx.x];
  __syncthreads();

  const int m = blockIdx.x * 256 + threadIdx.x;   // 0..32767
  const float* hrow = hs + (size_t)m * H_SZ;

  float l[4] = {bl[0], bl[1], bl[2], bl[3]};
#pragma unroll 4
  for (int h = 0; h < H_SZ; h += 4) {
    float4 v = *(const float4*)(hrow + h);
    float a;
    a = fmaxf(v.x, 0.f); l[0] += a * Wl[h+0]; l[1] += a * Wl[H_SZ+h+0]; l[2] += a * Wl[2*H_SZ+h+0]; l[3] += a * Wl[3*H_SZ+h+0];
    a = fmaxf(v.y, 0.f); l[0] += a * Wl[h+1]; l[1] += a * Wl[H_SZ+h+1]; l[2] += a * Wl[2*H_SZ+h+1]; l[3] += a * Wl[3*H_SZ+h+1];
    a = fmaxf(v.z, 0.f); l[0] += a * Wl[h+2]; l[1] += a * Wl[H_SZ+h+2]; l[2] += a * Wl[2*H_SZ+h+2]; l[3] += a * Wl[3*H_SZ+h+2];
    a = fmaxf(v.w, 0.f); l[0] += a * Wl[h+3]; l[1] += a * Wl[H_SZ+h+3]; l[2] += a * Wl[2*H_SZ+h+3]; l[3] += a * Wl[3*H_SZ+h+3];
  }

  float mx = fmaxf(fmaxf(l[0], l[1]), fmaxf(l[2], l[3]));
  float p[4], s = 0.f;
#pragma unroll
  for (int e = 0; e < 4; ++e) { p[e] = expf(l[e] - mx); s += p[e]; }
  float inv = 1.0f / s;

  const float cost[4] = {256.f, 512.f, 1024.f, 2048.f};
  float usage = 0.f;
#pragma unroll
  for (int e = 0; e < 4; ++e) {
    float pr = p[e] * inv;
    out_raw[(size_t)m * 4 + e]    = pr;
    out_router[(size_t)m * 4 + e] = pr;
    if (idx_is_i64) ((long long*)out_idx)[(size_t)m * 4 + e] = (long long)e;
    else            out_idx[(size_t)m * 4 + e] = (float)e;
    usage += pr * cost[e];
  }

  int tsk = task_ids[m] & 7;
  atomicAdd(&sums[tsk], usage);
  atomicAdd(&counts[tsk], 1.0f);
}

// ---------------------------------------------------------------------------
// Kernel 3c: losses[i] = sums[i] / counts[i]
// ---------------------------------------------------------------------------
__global__ void finalize_losses(const float* __restrict__ sums,
                                const float* __restrict__ counts,
                                float* __restrict__ losses) {
  if (threadIdx.x < 8) losses[threadIdx.x] = sums[threadIdx.x] / counts[threadIdx.x];
}

// ---------------------------------------------------------------------------
extern "C" void kernel_launch(void* const* d_in, const int* in_sizes, int n_in,
                              void* d_out, int out_size, void* d_ws, size_t ws_size,
                              hipStream_t stream) {
  const float* x    = (const float*)d_in[0];   // (S,B,D)
  const int*   tids = (const int*)  d_in[1];   // (S,B)
  const float* Wih  = (const float*)d_in[2];   // (3H,D)
  const float* Whh  = (const float*)d_in[3];   // (3H,H)
  const float* bih  = (const float*)d_in[4];   // (3H,)
  const float* bhh  = (const float*)d_in[5];   // (3H,)
  const float* Wout = (const float*)d_in[6];   // (E,H)
  const float* bout = (const float*)d_in[7];   // (E,)

  // workspace: xp (32768x768) | hs (32768x256) | sums(8) | counts(8)
  float* xp     = (float*)d_ws;
  float* hs     = xp + (size_t)M_ROWS * G3;
  float* sums   = hs + (size_t)M_ROWS * H_SZ;
  float* counts = sums + 8;

  float* out_raw    = (float*)d_out;
  float* out_router = out_raw + (size_t)M_ROWS * 4;
  float* out_idx    = out_router + (size_t)M_ROWS * 4;
  // indices are int64 in the reference; detect layout from out_size
  const int idx_i64 = (out_size == (int)(M_ROWS * 4 * 4 + 8)) ? 1 : 0;  // i64 = 2 f32 slots
  float* out_loss   = out_raw + (idx_i64 ? (size_t)M_ROWS * 16 : (size_t)M_ROWS * 12);

  // 1) big input-projection GEMM (parallel, WMMA f32, async-LDS double buffering)
  gemm_xproj<<<dim3(M_ROWS / 32, G3 / 384), 256, 0, stream>>>(x, Wih, bih, xp);
  // 2) sequential GRU scan (persistent single workgroup, WMMA f32)
  gru_scan<<<1, 512, 0, stream>>>(xp, Whh, bhh, hs);
  // 3) router head + fused segment reduction
  zero_bins<<<1, 64, 0, stream>>>(sums);
  router_head<<<M_ROWS / 256, 256, 0, stream>>>(hs, tids, Wout, bout,
                                                out_raw, out_router, out_idx,
                                                idx_i64, sums, counts);
  finalize_losses<<<1, 32, 0, stream>>>(sums, counts, out_loss);
}